// CurlVectorField_75892072120800
// MI455X (gfx1250) — compile-verified
//
#include <hip/hip_runtime.h>

#define NROWS 524288

typedef __attribute__((ext_vector_type(2))) float v2f;
typedef __attribute__((ext_vector_type(8))) float v8f;

__device__ __forceinline__ v8f wmma4(v2f a, v2f b, v8f c) {
  // D = A(16x4,f32) * B(4x16,f32) + C(16x16,f32)  -- exact fp32 matrix op
  return __builtin_amdgcn_wmma_f32_16x16x4_f32(false, a, false, b, (short)0, c,
                                               false, false);
}

__device__ __forceinline__ float tanh_fast(float v) {
#if __has_builtin(__builtin_amdgcn_tanhf)
  return __builtin_amdgcn_tanhf(v);          // gfx1250 v_tanh_f32
#elif __has_builtin(__builtin_amdgcn_tanh_f32)
  return __builtin_amdgcn_tanh_f32(v);
#else
  return tanhf(v);
#endif
}

// Intra-wave LDS fence: LDS ops are in-order per wave, so waiting DScnt==0
// makes prior ds_stores visible to later cross-lane ds_loads of this wave.
#define LDS_FENCE() asm volatile("s_wait_dscnt 0" ::: "memory")

__global__ __launch_bounds__(256) void curl_mlp_kernel(
    const float* __restrict__ x,  const float* __restrict__ W1,
    const float* __restrict__ b1, const float* __restrict__ W2,
    const float* __restrict__ b2, const float* __restrict__ W3,
    float* __restrict__ out)
{
  __shared__ float sW2[64 * 68];    // W2 row-major, padded stride 68
  __shared__ float sW1b[64 * 4];    // [n] = {W1[n][0..2], b1[n]}
  __shared__ float sb2[64];
  __shared__ float sGc[9 * 64];     // G_d[h][c] = eps(c,d,b)*W3[b][h], [d*3+c][h]
  __shared__ float sBuf[8 * 16 * 68]; // per-wave transpose buffer, stride 68

  const int tid = threadIdx.x;

  // ---- block-level weight staging ----
  for (int i = tid; i < 64 * 64; i += 256)
    sW2[(i >> 6) * 68 + (i & 63)] = W2[i];
  for (int i = tid; i < 64; i += 256) {
    sW1b[i * 4 + 0] = W1[i * 3 + 0];
    sW1b[i * 4 + 1] = W1[i * 3 + 1];
    sW1b[i * 4 + 2] = W1[i * 3 + 2];
    sW1b[i * 4 + 3] = b1[i];
    sb2[i] = b2[i];
  }
  for (int i = tid; i < 9 * 64; i += 256) {
    int k = i & 63, e = i >> 6;
    int dd = e / 3, cc = e % 3;
    float v = 0.0f;
    if (dd != cc) {
      int bsrc = 3 - dd - cc;                      // third index
      float s = (dd == ((cc + 1) % 3)) ? 1.0f : -1.0f;  // eps(c,d,b)
      v = s * W3[bsrc * 64 + k];
    }
    sGc[i] = v;
  }
  __syncthreads();

  const int lane = tid & 31;
  const int wv   = tid >> 5;
  const int l16  = lane & 15;
  const bool hi  = lane >= 16;
  const int hoff = hi ? 2 : 0;   // K offset of this lane-half in frags
  const int rsh  = hi ? 8 : 0;   // M offset of this lane-half in D tiles
  float* sA = sBuf + wv * (16 * 68);
  const long long r0 = (long long)(blockIdx.x * 8 + wv) * 16;

  // ---- layer 1: h1 = tanh([x|1] @ [W1^T; b1]) via K=4 WMMA ----
  v2f aX;
  {
    const float* xr = x + (r0 + l16) * 3;
    if (!hi) { aX.x = xr[0]; aX.y = xr[1]; }
    else     { aX.x = xr[2]; aX.y = 1.0f; }
  }
  v8f d1[4], d2[4];
  #pragma unroll
  for (int t = 0; t < 4; ++t) {
    v2f b = *(const v2f*)&sW1b[(16 * t + l16) * 4 + hoff];
    v8f c = {0, 0, 0, 0, 0, 0, 0, 0};
    c = wmma4(aX, b, c);
    #pragma unroll
    for (int r = 0; r < 8; ++r) {
      float h = tanh_fast(c[r]);
      d1[t][r] = 1.0f - h * h;                       // tanh' at layer 1
      sA[(r + rsh) * 68 + 16 * t + l16] = h;         // stage h1 for A-layout
    }
  }
  LDS_FENCE();

  v2f af[16];
  #pragma unroll
  for (int kk = 0; kk < 16; ++kk)
    af[kk] = *(const v2f*)&sA[l16 * 68 + 4 * kk + hoff];

  // ---- layer 2: h2 = tanh(h1 @ W2^T + b2); keep only D2 = 1 - h2^2 ----
  #pragma unroll
  for (int t = 0; t < 4; ++t) {
    float bias = sb2[16 * t + l16];
    v8f c = {bias, bias, bias, bias, bias, bias, bias, bias};
    #pragma unroll
    for (int kk = 0; kk < 16; ++kk) {
      v2f b = *(const v2f*)&sW2[(16 * t + l16) * 68 + 4 * kk + hoff];
      c = wmma4(af[kk], b, c);
    }
    #pragma unroll
    for (int r = 0; r < 8; ++r) {
      float h = tanh_fast(c[r]);
      d2[t][r] = 1.0f - h * h;
    }
  }

  // ---- Jacobian columns + fused curl contraction ----
  v8f curl = {0, 0, 0, 0, 0, 0, 0, 0};
  for (int d = 0; d < 3; ++d) {
    // T1_d = D1 * W1[:,d]  (elementwise scale in D-layout), stage to LDS
    #pragma unroll
    for (int t = 0; t < 4; ++t) {
      float w = sW1b[(16 * t + l16) * 4 + d];
      #pragma unroll
      for (int r = 0; r < 8; ++r)
        sA[(r + rsh) * 68 + 16 * t + l16] = d1[t][r] * w;
    }
    LDS_FENCE();
    #pragma unroll
    for (int kk = 0; kk < 16; ++kk)
      af[kk] = *(const v2f*)&sA[l16 * 68 + 4 * kk + hoff];

    // T2_d = D2 * (T1_d @ W2^T), stage to LDS
    #pragma unroll
    for (int t = 0; t < 4; ++t) {
      v8f m = {0, 0, 0, 0, 0, 0, 0, 0};
      #pragma unroll
      for (int kk = 0; kk < 16; ++kk) {
        v2f b = *(const v2f*)&sW2[(16 * t + l16) * 68 + 4 * kk + hoff];
        m = wmma4(af[kk], b, m);
      }
      #pragma unroll
      for (int r = 0; r < 8; ++r)
        sA[(r + rsh) * 68 + 16 * t + l16] = d2[t][r] * m[r];
    }
    LDS_FENCE();
    #pragma unroll
    for (int kk = 0; kk < 16; ++kk)
      af[kk] = *(const v2f*)&sA[l16 * 68 + 4 * kk + hoff];

    // curl += T2_d @ G_d   (G_d folds W3 with Levi-Civita; b3 drops out)
    #pragma unroll
    for (int kk = 0; kk < 16; ++kk) {
      v2f b = {0.0f, 0.0f};
      if (l16 < 3)
        b = *(const v2f*)&sGc[(d * 3 + l16) * 64 + 4 * kk + hoff];
      curl = wmma4(af[kk], b, curl);
    }
  }

  // ---- output: cols 0..2 of the curl D-tile -> 48 contiguous floats ----
  if (l16 < 3) {
    #pragma unroll
    for (int r = 0; r < 8; ++r)
      sA[(r + rsh) * 3 + l16] = curl[r];
  }
  LDS_FENCE();
  {
    float* o = out + r0 * 3;
    o[lane] = sA[lane];
    if (lane < 16) o[32 + lane] = sA[32 + lane];
  }
}

extern "C" void kernel_launch(void* const* d_in, const int* in_sizes, int n_in,
                              void* d_out, int out_size, void* d_ws, size_t ws_size,
                              hipStream_t stream) {
  (void)n_in; (void)out_size; (void)d_ws; (void)ws_size;
  const float* x  = (const float*)d_in[0];
  const float* W1 = (const float*)d_in[1];
  const float* b1 = (const float*)d_in[2];
  const float* W2 = (const float*)d_in[3];
  const float* b2 = (const float*)d_in[4];
  const float* W3 = (const float*)d_in[5];
  float* out = (float*)d_out;

  int rows = in_sizes[0] / 3;          // 524288; divisible by 128
  int blocks = rows / (16 * 8);        // 8 wave-tiles (16 rows) per 256-thr block
  curl_mlp_kernel<<<blocks, 256, 0, stream>>>(x, W1, b1, W2, b2, W3, out);
}